// DisulfideWholePoseScoringModule_83227876262359
// MI455X (gfx1250) — compile-verified
//
#include <hip/hip_runtime.h>
#include <hip/hip_bf16.h>

// Disulfide whole-pose scoring for MI455X (gfx1250, wave32).
// Gather/latency-bound (~65 MFLOP, ~10s of MB traffic @ 23.3 TB/s): the win is
// (1) LDS staging of the randomly re-indexed per-pose metadata rows via
//     gfx1250 async global->LDS loads (ASYNCcnt path),
// (2) early-out on invalid pairs,
// (3) trig-identity dihedral terms (no atan2f/cosf libcalls),
// (4) wave32 shuffle reduction.

#define P_      512
#define B_      512
#define APB_    32
#define A_      (B_ * APB_)
#define NBT_    8
#define MAXCONN_ 2
#define TPB     256
#define NWAVES  (TPB / 32)

#if defined(__has_builtin)
#  if __has_builtin(__builtin_amdgcn_global_load_async_to_lds_b32)
#    define USE_ASYNC_LDS 1
#  endif
#  if __has_builtin(__builtin_amdgcn_s_wait_asynccnt)
#    define HAVE_WAIT_ASYNC 1
#  endif
#endif

#if USE_ASYNC_LDS
// Parameter types per hipcc diagnostic: arg0 = addrspace(1) int* (global src),
// arg1 = addrspace(3) int* (LDS dst), then imm offset, imm cpol.
typedef __attribute__((address_space(1))) int g_int;
typedef __attribute__((address_space(3))) int l_int;
#endif

struct V3 { float x, y, z; };

__device__ __forceinline__ V3 v3(float a, float b, float c) { V3 r{a, b, c}; return r; }
__device__ __forceinline__ V3 vsub(V3 a, V3 b) { return v3(a.x - b.x, a.y - b.y, a.z - b.z); }
__device__ __forceinline__ float vdot(V3 a, V3 b) { return a.x * b.x + a.y * b.y + a.z * b.z; }
__device__ __forceinline__ V3 vcross(V3 a, V3 b) {
    return v3(a.y * b.z - a.z * b.y, a.z * b.x - a.x * b.z, a.x * b.y - a.y * b.x);
}
__device__ __forceinline__ V3 vunit(V3 a) {
    float inv = rsqrtf(vdot(a, a) + 1e-12f);
    return v3(a.x * inv, a.y * inv, a.z * inv);
}

// acos(clip(dot(unit(a-b), unit(c-b))))
__device__ __forceinline__ float angle3(V3 a, V3 b, V3 c) {
    V3 u = vunit(vsub(a, b));
    V3 w = vunit(vsub(c, b));
    float t = vdot(u, w);
    t = fminf(fmaxf(t, -1.0f + 1e-7f), 1.0f - 1e-7f);
    return acosf(t);
}

// dihedral(a,b,c,d) = atan2(y, x); we only need cos(2*phi) and cos(3*phi):
//   cos(phi)  = x / r,  r = sqrt(x^2 + y^2)
//   cos(2phi) = (x^2 - y^2) / r^2
//   cos(3phi) = cos(phi) * (4 cos^2(phi) - 3)
__device__ __forceinline__ void dihedral_xy(V3 a, V3 b, V3 c, V3 d, float& x, float& y) {
    V3 b1 = vsub(b, a);
    V3 b2 = vsub(c, b);
    V3 b3 = vsub(d, c);
    V3 n1 = vcross(b1, b2);
    V3 n2 = vcross(b2, b3);
    V3 m1 = vcross(n1, vunit(b2));
    x = vdot(n1, n2);
    y = vdot(m1, n2);
}

__device__ __forceinline__ V3 load_atom(const float* __restrict__ cp, int a) {
    const float* q = cp + 3 * (size_t)a;
    return v3(q[0], q[1], q[2]);
}

__global__ __launch_bounds__(TPB) void disulfide_score_kernel(
    const float* __restrict__ coords,   // [P][A][3]
    const int*   __restrict__ offs,     // [P][B]
    const int*   __restrict__ btypes,   // [P][B]
    const int*   __restrict__ conns,    // [P][B][MAXCONN][2]
    const int*   __restrict__ bt_dss,   // [NBT]
    const int*   __restrict__ bt_down,  // [NBT][MAXCONN][3]
    const float* __restrict__ gp,       // [6]
    float*       __restrict__ out)      // [P]
{
    __shared__ int   s_bt[B_];
    __shared__ int   s_off[B_];
    __shared__ float s_part[NWAVES];

    const int p   = blockIdx.x;
    const int tid = threadIdx.x;

    const int* bt_row  = btypes + (size_t)p * B_;
    const int* off_row = offs   + (size_t)p * B_;

    // ---- Stage per-pose metadata rows into LDS (randomly indexed by j later).
#if USE_ASYNC_LDS
    for (int k = tid; k < B_; k += TPB) {
        __builtin_amdgcn_global_load_async_to_lds_b32(
            (g_int*)(bt_row + k), (l_int*)(&s_bt[k]), 0, 0);
        __builtin_amdgcn_global_load_async_to_lds_b32(
            (g_int*)(off_row + k), (l_int*)(&s_off[k]), 0, 0);
    }
#  if HAVE_WAIT_ASYNC
    __builtin_amdgcn_s_wait_asynccnt(0);
#  else
    asm volatile("s_wait_asynccnt 0" ::: "memory");
#  endif
#else
    for (int k = tid; k < B_; k += TPB) {
        s_bt[k]  = bt_row[k];
        s_off[k] = off_row[k];
    }
#endif
    __syncthreads();

    // gfx1250 prefetch of this thread's second-trip connection record.
    {
        const int i2 = tid + TPB;
        if (i2 < B_)
            __builtin_prefetch(conns + (((size_t)p * B_ + i2) * MAXCONN_) * 2, 0, 1);
    }

    const float d0   = gp[0];
    const float wd   = gp[1];
    const float ang0 = gp[2];
    const float wa   = gp[3];
    const float wss  = gp[4];
    const float wcb  = gp[5];

    const float* cp = coords + (size_t)p * (A_ * 3);

    float acc = 0.0f;

    for (int i = tid; i < B_; i += TPB) {
        const int bt_i  = s_bt[i];
        const int dss_i = bt_dss[bt_i];
        if (dss_i < 0) continue;                 // has_i
        const int c_i = dss_i;

        const int* cn = conns + (((size_t)p * B_ + i) * MAXCONN_ + c_i) * 2;
        const int j  = cn[0];
        const int cj = cn[1];
        if (j < 0 || j <= i) continue;           // valid0 & (j > i)

        const int bt_j = s_bt[j];
        if (bt_dss[bt_j] != cj) continue;        // dss_j == cj

        const int* dni = bt_down + ((size_t)bt_i * MAXCONN_ + c_i) * 3;
        const int* dnj = bt_down + ((size_t)bt_j * MAXCONN_ + cj)  * 3;
        const int oi = s_off[i];
        const int oj = s_off[j];

        const V3 SGi = load_atom(cp, oi + dni[0]);
        const V3 CBi = load_atom(cp, oi + dni[1]);
        const V3 CAi = load_atom(cp, oi + dni[2]);
        const V3 SGj = load_atom(cp, oj + dnj[0]);
        const V3 CBj = load_atom(cp, oj + dnj[1]);
        const V3 CAj = load_atom(cp, oj + dnj[2]);

        // distance term
        V3 dv = vsub(SGi, SGj);
        float d  = sqrtf(vdot(dv, dv) + 1e-12f);
        float dd = d - d0;
        float E  = wd * dd * dd;

        // angle terms
        float a1 = angle3(CBi, SGi, SGj) - ang0;
        float a2 = angle3(CBj, SGj, SGi) - ang0;
        E += wa * (a1 * a1 + a2 * a2);

        // dihedral terms via cos(k*atan2(y,x)) identities
        float x1, y1, x2, y2, x3, y3;
        dihedral_xy(CBi, SGi, SGj, CBj, x1, y1);
        dihedral_xy(CAi, CBi, SGi, SGj, x2, y2);
        dihedral_xy(CAj, CBj, SGj, SGi, x3, y3);

        {   // wss * (1 + cos(2*t1))
            float r2 = x1 * x1 + y1 * y1 + 1e-30f;
            E += wss * (1.0f + (x1 * x1 - y1 * y1) / r2);
        }
        {   // wcb * (1 + cos(3*t2))
            float r2 = x2 * x2 + y2 * y2 + 1e-30f;
            float c  = x2 * rsqrtf(r2);
            E += wcb * (1.0f + c * (4.0f * c * c - 3.0f));
        }
        {   // wcb * (1 + cos(3*t3))
            float r2 = x3 * x3 + y3 * y3 + 1e-30f;
            float c  = x3 * rsqrtf(r2);
            E += wcb * (1.0f + c * (4.0f * c * c - 3.0f));
        }

        acc += E;
    }

    // ---- wave32 shuffle reduction, then cross-wave via LDS
    for (int o = 16; o > 0; o >>= 1)
        acc += __shfl_down(acc, o, 32);
    if ((tid & 31) == 0)
        s_part[tid >> 5] = acc;
    __syncthreads();
    if (tid == 0) {
        float t = 0.0f;
#pragma unroll
        for (int w = 0; w < NWAVES; ++w) t += s_part[w];
        out[p] = t;
    }
}

extern "C" void kernel_launch(void* const* d_in, const int* in_sizes, int n_in,
                              void* d_out, int out_size, void* d_ws, size_t ws_size,
                              hipStream_t stream) {
    const float* coords  = (const float*)d_in[0];
    const int*   offs    = (const int*)d_in[1];
    const int*   btypes  = (const int*)d_in[2];
    const int*   conns   = (const int*)d_in[3];
    const int*   bt_dss  = (const int*)d_in[4];
    const int*   bt_down = (const int*)d_in[5];
    const float* gp      = (const float*)d_in[6];
    float*       out     = (float*)d_out;

    (void)in_sizes; (void)n_in; (void)d_ws; (void)ws_size;

    dim3 grid((unsigned)out_size);   // one workgroup per pose (P = 512)
    disulfide_score_kernel<<<grid, TPB, 0, stream>>>(coords, offs, btypes, conns,
                                                     bt_dss, bt_down, gp, out);
}